// GCN_gen_17952963297475
// MI455X (gfx1250) — compile-verified
//
#include <hip/hip_runtime.h>
#include <stdint.h>

typedef __attribute__((ext_vector_type(2))) float v2f;
typedef __attribute__((ext_vector_type(8))) float v8f;

#define FEAT 64        // hidden/out channels
#define INC 128        // input channels

// ---------------------------------------------------------------------------
// Degree / normalization kernels
// ---------------------------------------------------------------------------
__global__ void init_deg_kernel(float* __restrict__ degW, float* __restrict__ deg1, int n) {
  int i = blockIdx.x * blockDim.x + threadIdx.x;
  if (i < n) { degW[i] = 1.0f; deg1[i] = 1.0f; }  // self-loop weight 1
}

__global__ void edge_deg_kernel(const long long* __restrict__ ei,
                                const float* __restrict__ ew,
                                float* __restrict__ degW, float* __restrict__ deg1, int E) {
  int e = blockIdx.x * blockDim.x + threadIdx.x;
  if (e >= E) return;
  int d = (int)ei[E + e];          // dst row of edge_index
  atomicAdd(&degW[d], ew[e]);
  atomicAdd(&deg1[d], 1.0f);
}

__global__ void rsqrt_kernel(float* __restrict__ degW, float* __restrict__ deg1, int n) {
  int i = blockIdx.x * blockDim.x + threadIdx.x;
  if (i >= n) return;
  float a = degW[i], b = deg1[i];
  degW[i] = (a > 0.0f) ? rsqrtf(a) : 0.0f;
  deg1[i] = (b > 0.0f) ? rsqrtf(b) : 0.0f;
}

// ---------------------------------------------------------------------------
// Dense transform H = X @ W via V_WMMA_F32_16X16X4_F32.
// One wave per 16x16 output tile. M, K multiples of 16/4; Nout = 64.
// A frag (16x4 f32): lane l: row = 16*mT + (l&15), holds K = kh, kh+1  (kh = (l>>4)*2)
// B frag (4x16 f32): lane l: col = 16*nT + (l&15), holds same K pair
// C/D (16x16 f32):   VGPR v, lane l: row = 16*mT + v + 8*(l>>4), col = 16*nT + (l&15)
// ---------------------------------------------------------------------------
__global__ void gemm_wmma_f32(const float* __restrict__ X, const float* __restrict__ W,
                              float* __restrict__ H, int M, int K, int Nout) {
  int wave = (int)((blockIdx.x * blockDim.x + threadIdx.x) >> 5);
  int lane = threadIdx.x & 31;
  int tilesN = Nout >> 4;
  int totalTiles = (M >> 4) * tilesN;
  if (wave >= totalTiles) return;               // wave-uniform exit, EXEC stays all-1s inside
  int mT = wave / tilesN;
  int nT = wave - mT * tilesN;

  int row = mT * 16 + (lane & 15);
  int col = nT * 16 + (lane & 15);
  int kh  = (lane >> 4) * 2;

  v8f acc = {};
  const float* xr = X + (size_t)row * K;
  for (int k0 = 0; k0 < K; k0 += 4) {
    v2f a, b;
    a.x = xr[k0 + kh];
    a.y = xr[k0 + kh + 1];
    b.x = W[(size_t)(k0 + kh) * Nout + col];
    b.y = W[(size_t)(k0 + kh + 1) * Nout + col];
    acc = __builtin_amdgcn_wmma_f32_16x16x4_f32(false, a, false, b,
                                                (short)0, acc, false, false);
  }

  int rbase = mT * 16 + 8 * (lane >> 4);
  int c = nT * 16 + (lane & 15);
#pragma unroll
  for (int v = 0; v < 8; ++v)
    H[(size_t)(rbase + v) * Nout + c] = acc[v];
}

// ---------------------------------------------------------------------------
// Self-loop contribution: AGG[i][c] = dinv[i]^2 * H[i][c]   (norm = dinv*1*dinv)
// ---------------------------------------------------------------------------
__global__ void self_init_kernel(const float* __restrict__ H, const float* __restrict__ dinv,
                                 float* __restrict__ agg, int total) {
  int i = blockIdx.x * blockDim.x + threadIdx.x;
  if (i >= total) return;
  float di = dinv[i >> 6];
  agg[i] = H[i] * di * di;
}

// ---------------------------------------------------------------------------
// Edge scatter: one wave per edge; lane covers 2 of the 64 features.
// AGG[dst][c] += H[src][c] * dinv[src]*w*dinv[dst]  (L2-resident atomics)
// ---------------------------------------------------------------------------
__global__ void edge_scatter_kernel(const long long* __restrict__ ei,
                                    const float* __restrict__ ew,
                                    const float* __restrict__ dinv,
                                    const float* __restrict__ H,
                                    float* __restrict__ agg,
                                    int E, int useEw) {
  int wave = (int)((blockIdx.x * blockDim.x + threadIdx.x) >> 5);
  int lane = threadIdx.x & 31;
  if (wave >= E) return;
  int s = (int)ei[wave];
  int d = (int)ei[E + wave];
  float w = useEw ? ew[wave] : 1.0f;
  float norm = dinv[s] * w * dinv[d];
  const float* hs = H + (size_t)s * FEAT;
  float* od = agg + (size_t)d * FEAT;
  int c = lane * 2;
  float m0 = hs[c] * norm;
  float m1 = hs[c + 1] * norm;
  atomicAdd(&od[c], m0);
  atomicAdd(&od[c + 1], m1);
}

// ---------------------------------------------------------------------------
// out = (agg + bias), optional ReLU
// ---------------------------------------------------------------------------
__global__ void bias_act_kernel(const float* __restrict__ agg, const float* __restrict__ bias,
                                float* __restrict__ out, int total, int doRelu) {
  int i = blockIdx.x * blockDim.x + threadIdx.x;
  if (i >= total) return;
  float v = agg[i] + bias[i & (FEAT - 1)];
  out[i] = doRelu ? fmaxf(v, 0.0f) : v;
}

// ---------------------------------------------------------------------------
// Host-side orchestration
// ---------------------------------------------------------------------------
extern "C" void kernel_launch(void* const* d_in, const int* in_sizes, int n_in,
                              void* d_out, int out_size, void* d_ws, size_t ws_size,
                              hipStream_t stream) {
  const float*     x  = (const float*)d_in[0];      // [N,128]
  const long long* ei = (const long long*)d_in[1];  // [2,E] int64
  const float*     ew = (const float*)d_in[2];      // [E]
  const float*     W1 = (const float*)d_in[3];      // [128,64]
  const float*     b1 = (const float*)d_in[4];
  const float*     W2 = (const float*)d_in[5];      // [64,64]
  const float*     b2 = (const float*)d_in[6];
  const float*     W3 = (const float*)d_in[7];      // [64,64]
  const float*     b3 = (const float*)d_in[8];

  const int N = in_sizes[0] / INC;   // 50000
  const int E = in_sizes[2];         // 800000
  const int NF = N * FEAT;

  // Workspace layout (floats)
  float* dinvW = (float*)d_ws;          // [N]
  float* dinv1 = dinvW + N;             // [N]
  float* H     = dinv1 + N;             // [N,64] GEMM output
  float* AGG   = H + NF;                // [N,64] aggregation accumulator
  float* ACT   = AGG + NF;              // [N,64] activated (next-layer input)
  float* outF  = (float*)d_out;

  const int TB = 256;
  dim3 blk(TB);
  int gN   = (N + TB - 1) / TB;
  int gE   = (E + TB - 1) / TB;
  int gNF  = (NF + TB - 1) / TB;
  int gEw  = (int)(((long long)E * 32 + TB - 1) / TB);   // one wave per edge
  auto gemmGrid = [&](int M, int Nout) {
    long long waves = (long long)(M / 16) * (Nout / 16);
    return (int)((waves * 32 + TB - 1) / TB);
  };

  // --- normalization coefficients (weighted and unit-weight variants) ---
  init_deg_kernel<<<gN, blk, 0, stream>>>(dinvW, dinv1, N);
  edge_deg_kernel<<<gE, blk, 0, stream>>>(ei, ew, dinvW, dinv1, E);
  rsqrt_kernel<<<gN, blk, 0, stream>>>(dinvW, dinv1, N);

  // --- layer 1: X[N,128] @ W1 -> relu(agg + b1) ---
  gemm_wmma_f32<<<gemmGrid(N, FEAT), blk, 0, stream>>>(x, W1, H, N, INC, FEAT);
  self_init_kernel<<<gNF, blk, 0, stream>>>(H, dinvW, AGG, NF);
  edge_scatter_kernel<<<gEw, blk, 0, stream>>>(ei, ew, dinvW, H, AGG, E, 1);
  bias_act_kernel<<<gNF, blk, 0, stream>>>(AGG, b1, ACT, NF, 1);

  // --- layer 2: ACT @ W2 -> relu(agg + b2) ---
  gemm_wmma_f32<<<gemmGrid(N, FEAT), blk, 0, stream>>>(ACT, W2, H, N, FEAT, FEAT);
  self_init_kernel<<<gNF, blk, 0, stream>>>(H, dinvW, AGG, NF);
  edge_scatter_kernel<<<gEw, blk, 0, stream>>>(ei, ew, dinvW, H, AGG, E, 1);
  bias_act_kernel<<<gNF, blk, 0, stream>>>(AGG, b2, ACT, NF, 1);

  // --- layer 3: ACT @ W3, unit edge weights, no ReLU, straight to d_out ---
  gemm_wmma_f32<<<gemmGrid(N, FEAT), blk, 0, stream>>>(ACT, W3, H, N, FEAT, FEAT);
  self_init_kernel<<<gNF, blk, 0, stream>>>(H, dinv1, AGG, NF);
  edge_scatter_kernel<<<gEw, blk, 0, stream>>>(ei, ew, dinv1, H, AGG, E, 0);
  bias_act_kernel<<<gNF, blk, 0, stream>>>(AGG, b3, outF, NF, 0);
}